// Top1Gate_33578054320708
// MI455X (gfx1250) — compile-verified
//
#include <hip/hip_runtime.h>
#include <math.h>

#define N_TOK 65536
#define MDIM  1024
#define NEXP  64
#define BM    128
#define BK    64
#define XSTR  68          // LDS row stride (floats): 68%64=4 -> conflict-free b64 frag reads
#define WSTR  68
#define LSTR  65          // logits LDS stride
#define NC    (MDIM / BK)    // 16 K-chunks
#define NBLK  (N_TOK / BM)   // 512
#define CHUNK_F (BM * XSTR + NEXP * WSTR)   // floats per LDS buffer (13056)

typedef __attribute__((ext_vector_type(2))) float v2f;
typedef __attribute__((ext_vector_type(8))) float v8f;
typedef __attribute__((ext_vector_type(4))) int   v4i;

typedef __attribute__((address_space(3))) v4i* lds_v4i_p;   // LDS-typed b128 pointer

// ---------------------------------------------------------------------------
// Kernel 1: fused gemm (logits = x @ wg^T) + per-token argmax/softmax +
// per-block deterministic partials (me, expert histogram).
// 512 blocks x 256 threads (8 wave32); each wave owns a 16x64 logit tile.
// Double-buffered GLOBAL_LOAD_ASYNC_TO_LDS_B128 staging (ASYNCcnt-tracked)
// overlaps the copy engine with the WMMA pipeline.
// ---------------------------------------------------------------------------
__global__ __launch_bounds__(256) void gate_gemm_kernel(
    const float* __restrict__ x, const float* __restrict__ wg,
    float* __restrict__ out, int* __restrict__ expert_ids,
    int* __restrict__ hist, float* __restrict__ me_part)
{
  __shared__ float smem[2 * CHUNK_F];     // ~104 KB: two staging buffers
  __shared__ float lmax[BM];
  __shared__ float linv[BM];
  __shared__ int   lexp[BM];

  const int  tid  = threadIdx.x;
  const int  blk  = blockIdx.x;
  const int  wave = tid >> 5;
  const int  lane = tid & 31;
  const int  half = lane >> 4;       // K sub-group for 16x4 f32 fragments
  const int  ln   = lane & 15;
  const long rowbase = (long)blk * BM;

  // Issue one K-chunk's async copies into `buf` (12 b128 ops per thread).
  auto issue_chunk = [&](int kc, float* buf) {
    const int K0 = kc * BK;
    #pragma unroll
    for (int i = 0; i < 8; ++i) {            // x tile: 128x64 f32 = 2048 float4
      int f  = tid + 256 * i;
      int r  = f >> 4;                       // 16 float4 per row
      int c4 = (f & 15) << 2;
      __builtin_amdgcn_global_load_async_to_lds_b128(
          (v4i*)&x[(rowbase + r) * MDIM + K0 + c4],
          (lds_v4i_p)&buf[r * XSTR + c4], 0, 0);
    }
    #pragma unroll
    for (int i = 0; i < 4; ++i) {            // wg tile: 64x64 f32 = 1024 float4
      int f  = tid + 256 * i;
      int e  = f >> 4;
      int c4 = (f & 15) << 2;
      __builtin_amdgcn_global_load_async_to_lds_b128(
          (v4i*)&wg[(long)e * MDIM + K0 + c4],
          (lds_v4i_p)&buf[BM * XSTR + e * WSTR + c4], 0, 0);
    }
  };

  v8f c0 = {}; v8f c1 = {}; v8f c2 = {}; v8f c3 = {};

  issue_chunk(0, smem);
  for (int kc = 0; kc < NC; ++kc) {
    float* cur = smem + (kc & 1) * CHUNK_F;
    if (kc + 1 < NC) {
      issue_chunk(kc + 1, smem + ((kc + 1) & 1) * CHUNK_F);
      __builtin_amdgcn_s_wait_asynccnt(12);  // retire chunk kc, keep kc+1 in flight
    } else {
      __builtin_amdgcn_s_wait_asynccnt(0);
    }
    __syncthreads();                          // all waves' LDS writes visible

    float* xs  = cur;
    float* wsm = cur + BM * XSTR;
    const int arow = (wave << 4) + ln;
    #pragma unroll
    for (int kk = 0; kk < BK; kk += 4) {
      const int ko = kk + (half << 1);
      v2f a  = *(const v2f*)&xs [arow * XSTR + ko];
      v2f b0 = *(const v2f*)&wsm[(0 * 16 + ln) * WSTR + ko];
      v2f b1 = *(const v2f*)&wsm[(1 * 16 + ln) * WSTR + ko];
      v2f b2 = *(const v2f*)&wsm[(2 * 16 + ln) * WSTR + ko];
      v2f b3 = *(const v2f*)&wsm[(3 * 16 + ln) * WSTR + ko];
      c0 = __builtin_amdgcn_wmma_f32_16x16x4_f32(false, a, false, b0, (short)0, c0, false, false);
      c1 = __builtin_amdgcn_wmma_f32_16x16x4_f32(false, a, false, b1, (short)0, c1, false, false);
      c2 = __builtin_amdgcn_wmma_f32_16x16x4_f32(false, a, false, b2, (short)0, c2, false, false);
      c3 = __builtin_amdgcn_wmma_f32_16x16x4_f32(false, a, false, b3, (short)0, c3, false, false);
    }
    __syncthreads();                          // buffer reusable for chunk kc+2
  }

  // ---- spill logits tile to LDS: Lg[row 0..127][expert 0..63], stride 65 ----
  float* Lg = smem;
  {
    const int rbase = (wave << 4) + (half << 3);   // C layout: row = r + 8*half
    #pragma unroll
    for (int r = 0; r < 8; ++r) {
      Lg[(rbase + r) * LSTR + 0 * 16 + ln] = c0[r];
      Lg[(rbase + r) * LSTR + 1 * 16 + ln] = c1[r];
      Lg[(rbase + r) * LSTR + 2 * 16 + ln] = c2[r];
      Lg[(rbase + r) * LSTR + 3 * 16 + ln] = c3[r];
    }
  }
  __syncthreads();

  // ---- per-token argmax + softmax (first-max tie rule like jnp.argmax) ----
  if (tid < BM) {
    const float* row = &Lg[tid * LSTR];
    float m = row[0]; int am = 0;
    #pragma unroll 4
    for (int e = 1; e < NEXP; ++e) { float v = row[e]; if (v > m) { m = v; am = e; } }
    float s = 0.f;
    #pragma unroll 4
    for (int e = 0; e < NEXP; ++e) s += __expf(row[e] - m);
    const float inv = 1.f / s;       // softmax prob at argmax
    lmax[tid] = m; linv[tid] = inv; lexp[tid] = am;
    const long g = rowbase + tid;
    out[1 + g]              = (float)am;   // indices1_s
    out[2L * N_TOK + 2 + g] = inv;         // gates1_s
    expert_ids[g]           = am;
  }
  __syncthreads();

  // ---- deterministic per-expert block partials (no float atomics) ----
  if (tid < NEXP) {
    const int e = tid;
    float ms = 0.f; int cnt = 0;
    for (int j = 0; j < BM; ++j) {
      ms  += __expf(Lg[j * LSTR + e] - lmax[j]) * linv[j];
      cnt += (lexp[j] == e) ? 1 : 0;
    }
    me_part[blk * NEXP + e] = ms;
    hist   [blk * NEXP + e] = cnt;
  }
}

// ---------------------------------------------------------------------------
// Kernel 2: per-expert exclusive scan over block histograms + l_aux + scalars
// ---------------------------------------------------------------------------
__global__ __launch_bounds__(64) void scan_kernel(
    const int* __restrict__ hist, const float* __restrict__ me_part,
    int* __restrict__ offsets, float* __restrict__ out)
{
  __shared__ float red[NEXP];
  const int e = threadIdx.x;     // 64 threads, one per expert
  int run = 0; float ms = 0.f;
  for (int b = 0; b < NBLK; ++b) {
    offsets[b * NEXP + e] = run;
    run += hist[b * NEXP + e];
    ms  += me_part[b * NEXP + e];
  }
  red[e] = ms * (float)run;      // me[e] * ce[e]
  __syncthreads();
  if (e == 0) {
    float t = 0.f;
    for (int i = 0; i < NEXP; ++i) t += red[i];
    out[0]              = t * ((float)NEXP / ((float)N_TOK * (float)N_TOK)); // l_aux
    out[1 + N_TOK]      = (float)((N_TOK + NEXP - 1) / NEXP);               // capacity
    out[3L * N_TOK + 2] = (float)NEXP;                                      // E
  }
}

// ---------------------------------------------------------------------------
// Kernel 3: locations1_s = block-offset[expert] + within-block ordered rank
// ---------------------------------------------------------------------------
__global__ __launch_bounds__(128) void loc_kernel(
    const int* __restrict__ expert_ids, const int* __restrict__ offsets,
    float* __restrict__ out)
{
  __shared__ int le[BM];
  const int  tid = threadIdx.x;
  const int  blk = blockIdx.x;
  const long g   = (long)blk * BM + tid;
  le[tid] = expert_ids[g];
  __syncthreads();
  const int e = le[tid];
  int cnt = 0;
  for (int j = 0; j < tid; ++j) cnt += (le[j] == e) ? 1 : 0;
  out[(long)N_TOK + 2 + g] = (float)(offsets[blk * NEXP + e] + cnt);
}

// ---------------------------------------------------------------------------
extern "C" void kernel_launch(void* const* d_in, const int* in_sizes, int n_in,
                              void* d_out, int out_size, void* d_ws, size_t ws_size,
                              hipStream_t stream) {
  const float* x  = (const float*)d_in[0];   // [65536,1024]
  const float* wg = (const float*)d_in[1];   // [64,1024]
  float* out = (float*)d_out;                // 3*N+3 floats
  char* ws = (char*)d_ws;
  int*   expert_ids = (int*)  (ws);                 // 256 KB
  int*   hist       = (int*)  (ws + 256 * 1024);    // 128 KB
  int*   offsets    = (int*)  (ws + 384 * 1024);    // 128 KB
  float* me_part    = (float*)(ws + 512 * 1024);    // 128 KB

  gate_gemm_kernel<<<NBLK, 256, 0, stream>>>(x, wg, out, expert_ids, hist, me_part);
  scan_kernel<<<1, NEXP, 0, stream>>>(hist, me_part, offsets, out);
  loc_kernel<<<NBLK, BM, 0, stream>>>(expert_ids, offsets, out);
}